// Basic_Net_29721173689135
// MI455X (gfx1250) — compile-verified
//
#include <hip/hip_runtime.h>
#include <math.h>

typedef float v2f __attribute__((ext_vector_type(2)));
typedef float v8f __attribute__((ext_vector_type(8)));

#define N_NODES 100000
#define N_EDGES 3200000
#define F_IN    32
#define HID     64

// ---------------- utility ----------------
__global__ void k_zero(float* __restrict__ p, long n) {
  long i = (long)blockIdx.x * blockDim.x + threadIdx.x;
  if (i < n) p[i] = 0.0f;
}

// ---------------- degree ----------------
__global__ void k_degree(const int* __restrict__ col, float* __restrict__ deg, int E) {
  int e = blockIdx.x * blockDim.x + threadIdx.x;
  if (e < E) atomicAdd(&deg[col[e]], 1.0f);
}

__global__ void k_deg_fin(const float* __restrict__ deg, float* __restrict__ disq,
                          float* __restrict__ dinv, int n) {
  int i = blockIdx.x * blockDim.x + threadIdx.x;
  if (i < n) {
    float d = deg[i] + 1.0f;       // self-loop
    float r = rsqrtf(d);
    disq[i] = r;
    dinv[i] = r * r;               // == 1/deg
  }
}

// ---------------- label sum (n_pos) ----------------
__global__ void k_label_sum(const float* __restrict__ labels, float* __restrict__ npos, int n) {
  __shared__ float red[256];
  int i = blockIdx.x * blockDim.x + threadIdx.x;
  float v = (i < n) ? labels[i] : 0.0f;
  red[threadIdx.x] = v;
  __syncthreads();
  for (int s = 128; s > 0; s >>= 1) {
    if (threadIdx.x < s) red[threadIdx.x] += red[threadIdx.x + s];
    __syncthreads();
  }
  if (threadIdx.x == 0) atomicAdd(npos, red[0]);
}

// ---------------- dense GEMM via V_WMMA_F32_16X16X4_F32 ----------------
// M = A[N,K] * W[K,64].  One wave computes one 16x16 output tile.
// A fragment (16x4, MxK): lanes 0-15 -> M=l16, K = k+0/k+1 in the 2 VGPRs;
//                         lanes16-31 -> M=l16, K = k+2/k+3.
// B fragment (4x16, KxN): VGPR0: lanes0-15 K=k+0, lanes16-31 K=k+2;
//                         VGPR1: lanes0-15 K=k+1, lanes16-31 K=k+3.
// C/D (16x16 f32): VGPR r: lanes0-15 M=r, lanes16-31 M=8+r; N = lane&15.
__global__ void k_gemm_wmma(const float* __restrict__ A, const float* __restrict__ W,
                            float* __restrict__ Mo, int K) {
  int gw   = blockIdx.x * (blockDim.x >> 5) + (threadIdx.x >> 5);
  int lane = threadIdx.x & 31;
  int nt   = gw & 3;          // 64 cols = 4 tiles of 16
  int mt   = gw >> 2;         // 6250 row tiles (100000/16 exactly)
  int half = lane >> 4;
  int l16  = lane & 15;

  const float* Ar = A + (size_t)(mt * 16 + l16) * K;
  const float* Wc = W + nt * 16 + l16;

  v8f c = {0.f, 0.f, 0.f, 0.f, 0.f, 0.f, 0.f, 0.f};
  for (int k = 0; k < K; k += 4) {
    v2f a, b;
    a.x = Ar[k + 2 * half + 0];
    a.y = Ar[k + 2 * half + 1];
    b.x = Wc[(size_t)(k + 2 * half + 0) * HID];
    b.y = Wc[(size_t)(k + 2 * half + 1) * HID];
    c = __builtin_amdgcn_wmma_f32_16x16x4_f32(
        /*neg_a=*/false, a, /*neg_b=*/false, b,
        /*c_mod=*/(short)0, c, /*reuse_a=*/false, /*reuse_b=*/false);
  }

  float* Or = Mo + (size_t)(mt * 16 + half * 8) * HID + nt * 16 + l16;
#pragma unroll
  for (int r = 0; r < 8; ++r) Or[(size_t)r * HID] = c[r];
}

// ---------------- edge scatter: agg[col] += m[row] * disq[row]*disq[col] --------
// 16 threads per edge, float4 (b128) gather + 4 f32 atomics (resolve in L2).
__global__ void k_scatter(const float* __restrict__ m, const int* __restrict__ row,
                          const int* __restrict__ col, const float* __restrict__ disq,
                          float* __restrict__ agg, int E) {
  long t = (long)blockIdx.x * blockDim.x + threadIdx.x;
  long e = t >> 4;
  int  ch = (int)(t & 15);
  if (e >= E) return;
  // stream-prefetch upcoming edge indices (emits global_prefetch_b8)
  long pe = e + 4096; if (pe >= E) pe = E - 1;
  __builtin_prefetch(&row[pe], 0, 1);
  __builtin_prefetch(&col[pe], 0, 1);

  int r = row[e], c = col[e];
  float coef = disq[r] * disq[c];
  const float4 v = *(const float4*)(m + (size_t)r * HID + ch * 4);
  float* dst = agg + (size_t)c * HID + ch * 4;
  atomicAdd(dst + 0, v.x * coef);
  atomicAdd(dst + 1, v.y * coef);
  atomicAdd(dst + 2, v.z * coef);
  atomicAdd(dst + 3, v.w * coef);
}

// ---------------- self-loop + bias + relu ----------------
__global__ void k_finalize(const float* __restrict__ m, const float* __restrict__ agg,
                           const float* __restrict__ dinv, const float* __restrict__ b,
                           float* __restrict__ hout, int n) {
  long t = (long)blockIdx.x * blockDim.x + threadIdx.x;
  if (t >= (long)n * 16) return;
  int i  = (int)(t >> 4);
  int ch = (int)(t & 15) * 4;
  float di = dinv[i];
  const float4 mv = *(const float4*)(m   + (size_t)i * HID + ch);
  const float4 av = *(const float4*)(agg + (size_t)i * HID + ch);
  const float4 bv = *(const float4*)(b + ch);
  float4 o;
  o.x = fmaxf(av.x + mv.x * di + bv.x, 0.0f);
  o.y = fmaxf(av.y + mv.y * di + bv.y, 0.0f);
  o.z = fmaxf(av.z + mv.z * di + bv.z, 0.0f);
  o.w = fmaxf(av.w + mv.w * di + bv.w, 0.0f);
  *(float4*)(hout + (size_t)i * HID + ch) = o;
}

// ---------------- MLP head + sigmoid + weighted BCE partial sums ----------------
__global__ void k_head(const float* __restrict__ h, const float* __restrict__ Wl1,
                       const float* __restrict__ bl1, const float* __restrict__ Wl2,
                       const float* __restrict__ bl2, const float* __restrict__ labels,
                       const float* __restrict__ npos_ptr, float* __restrict__ p_out,
                       float* __restrict__ bce_sum, int n) {
  __shared__ float red[256];
  int i = blockIdx.x * blockDim.x + threadIdx.x;
  float contrib = 0.0f;
  if (i < n) {
    const float* hr = h + (size_t)i * HID;
    float acc[8];
#pragma unroll
    for (int j = 0; j < 8; ++j) acc[j] = bl1[j];
    for (int f = 0; f < HID; ++f) {
      float hv = hr[f];
#pragma unroll
      for (int j = 0; j < 8; ++j) acc[j] = fmaf(hv, Wl1[f * 8 + j], acc[j]);
    }
    float z = bl2[0];
#pragma unroll
    for (int j = 0; j < 8; ++j) z = fmaf(fmaxf(acc[j], 0.0f), Wl2[j], z);

    float pv = 1.0f / (1.0f + __expf(-z));
    p_out[i] = pv;

    float y = labels[i];
    // stable BCE on logits: softplus(z) - y*z
    float bce = fmaxf(z, 0.0f) - z * y + __logf(1.0f + __expf(-fabsf(z)));
    float npos = *npos_ptr;
    float ntot = (float)n;
    float wpos = ntot / (2.0f * npos);
    float wneg = ntot / (2.0f * (ntot - npos));
    float w = y * wpos + (1.0f - y) * wneg;
    contrib = w * bce;
  }
  red[threadIdx.x] = contrib;
  __syncthreads();
  for (int s = 128; s > 0; s >>= 1) {
    if (threadIdx.x < s) red[threadIdx.x] += red[threadIdx.x + s];
    __syncthreads();
  }
  if (threadIdx.x == 0) atomicAdd(bce_sum, red[0]);
}

__global__ void k_loss(const float* __restrict__ bce_sum, float* __restrict__ loss, int n) {
  if (threadIdx.x == 0 && blockIdx.x == 0) loss[0] = bce_sum[0] / (float)n;
}

// ---------------- launch ----------------
extern "C" void kernel_launch(void* const* d_in, const int* in_sizes, int n_in,
                              void* d_out, int out_size, void* d_ws, size_t ws_size,
                              hipStream_t stream) {
  const float* x      = (const float*)d_in[0];
  const int*   ei     = (const int*)d_in[1];
  const int*   row    = ei;
  const int*   col    = ei + N_EDGES;
  const float* labels = (const float*)d_in[2];
  const float* W1 = (const float*)d_in[3];  const float* b1 = (const float*)d_in[4];
  const float* W2 = (const float*)d_in[5];  const float* b2 = (const float*)d_in[6];
  const float* W3 = (const float*)d_in[7];  const float* b3 = (const float*)d_in[8];
  const float* Wl1 = (const float*)d_in[9]; const float* bl1 = (const float*)d_in[10];
  const float* Wl2 = (const float*)d_in[11];const float* bl2 = (const float*)d_in[12];

  float* ws   = (float*)d_ws;
  const size_t NH = (size_t)N_NODES * HID;
  float* m    = ws;                 // [N,64]
  float* agg  = ws + NH;            // [N,64]
  float* hbuf = ws + 2 * NH;        // [N,64]
  float* deg  = ws + 3 * NH;        // [N]
  float* disq = deg + N_NODES;      // [N]
  float* dinv = disq + N_NODES;     // [N]
  float* scal = dinv + N_NODES;     // [0]=n_pos, [1]=bce_sum

  float* loss_out = (float*)d_out;
  float* p_out    = loss_out + 1;

  const int T = 256;
  const int blocksN   = (N_NODES + T - 1) / T;          // 391
  const int blocksE   = (N_EDGES + T - 1) / T;          // 12500
  const int blocksNH  = (int)((NH + T - 1) / T);        // 25000
  const int blocksN16 = (int)(((long)N_NODES * 16 + T - 1) / T);  // 6250
  const int blocksE16 = (int)(((long)N_EDGES * 16 + T - 1) / T);  // 200000
  const int blocksGEMM = (N_NODES / 16) * (HID / 16) / (T / 32);  // 3125

  // degree + normalization + class-balance sums
  k_zero<<<blocksN, T, 0, stream>>>(deg, (long)N_NODES);
  k_zero<<<1, T, 0, stream>>>(scal, 2L);
  k_degree<<<blocksE, T, 0, stream>>>(col, deg, N_EDGES);
  k_deg_fin<<<blocksN, T, 0, stream>>>(deg, disq, dinv, N_NODES);
  k_label_sum<<<blocksN, T, 0, stream>>>(labels, &scal[0], N_NODES);

  // GCN layer 1 (in = x, K=32)
  k_gemm_wmma<<<blocksGEMM, T, 0, stream>>>(x, W1, m, F_IN);
  k_zero<<<blocksNH, T, 0, stream>>>(agg, (long)NH);
  k_scatter<<<blocksE16, T, 0, stream>>>(m, row, col, disq, agg, N_EDGES);
  k_finalize<<<blocksN16, T, 0, stream>>>(m, agg, dinv, b1, hbuf, N_NODES);

  // GCN layer 2 (in = hbuf, K=64)
  k_gemm_wmma<<<blocksGEMM, T, 0, stream>>>(hbuf, W2, m, HID);
  k_zero<<<blocksNH, T, 0, stream>>>(agg, (long)NH);
  k_scatter<<<blocksE16, T, 0, stream>>>(m, row, col, disq, agg, N_EDGES);
  k_finalize<<<blocksN16, T, 0, stream>>>(m, agg, dinv, b2, hbuf, N_NODES);

  // GCN layer 3 (in = hbuf, K=64)
  k_gemm_wmma<<<blocksGEMM, T, 0, stream>>>(hbuf, W3, m, HID);
  k_zero<<<blocksNH, T, 0, stream>>>(agg, (long)NH);
  k_scatter<<<blocksE16, T, 0, stream>>>(m, row, col, disq, agg, N_EDGES);
  k_finalize<<<blocksN16, T, 0, stream>>>(m, agg, dinv, b3, hbuf, N_NODES);

  // head + loss
  k_head<<<blocksN, T, 0, stream>>>(hbuf, Wl1, bl1, Wl2, bl2, labels,
                                    &scal[0], p_out, &scal[1], N_NODES);
  k_loss<<<1, 64, 0, stream>>>(&scal[1], loss_out, N_NODES);
}